// GCN_SAGPool_27161373180173
// MI455X (gfx1250) — compile-verified
//
#include <hip/hip_runtime.h>
#include <hip/hip_bf16.h>
#include <math.h>

typedef __attribute__((ext_vector_type(2))) float v2f;
typedef __attribute__((ext_vector_type(8))) float v8f;

static constexpr int NN  = 100000;   // nodes
static constexpr int EE  = 3200000;  // edges
static constexpr int FF  = 128;      // input features
static constexpr int HH  = 16;       // hidden
static constexpr int CC  = 10;       // classes
static constexpr int KK1 = 25000;    // pool1 keep
static constexpr int KK2 = 6250;     // pool2 keep

__device__ __forceinline__ void atomAddF(float* p, float v) {
  unsafeAtomicAdd(p, v);  // global_atomic_add_f32
}

// ---------------- generic fills ----------------
__global__ void k_fill_f32(float* p, float v, int n) {
  int i = blockIdx.x * blockDim.x + threadIdx.x;
  if (i < n) p[i] = v;
}
__global__ void k_fill_i32(int* p, int v, int n) {
  int i = blockIdx.x * blockDim.x + threadIdx.x;
  if (i < n) p[i] = v;
}

// ---------------- WMMA f32 GEMM: D[Mx16] = A[MxK] * B[K x NCOLS->pad16] ----
// One wave per 16-row tile, compile-time K -> fully unrolled WMMA chain.
// Branchless: addresses clamped, values masked, loads unconditional.
// A 16x4 f32 operand: lanes 0-15 hold K=k0,k0+1; lanes 16-31 K=k0+2,k0+3.
// B 4x16 mirrored by N.  C/D: lane = column, vgpr v -> row 8*(lane/16)+v.
template <int K, int LDB, int NCOLS>
__global__ void k_gemm16(const float* __restrict__ A, const float* __restrict__ B,
                         float* __restrict__ D, int M) {
  int wave = blockIdx.x * (blockDim.x >> 5) + (threadIdx.x >> 5);
  int lane = threadIdx.x & 31;
  int ntiles = (M + 15) >> 4;
  if (wave >= ntiles) return;                 // wave-uniform exit
  int rrow = (wave << 4) + (lane & 15);
  int row  = min(rrow, M - 1);                // clamped, always in-bounds
  float am = (rrow < M) ? 1.f : 0.f;          // zero tail rows via multiply
  int ccol = lane & 15;
  int col  = (NCOLS < 16) ? min(ccol, NCOLS - 1) : ccol;
  float bm = (NCOLS < 16) ? ((ccol < NCOLS) ? 1.f : 0.f) : 1.f;
  int khalf = (lane >> 4) << 1;               // 0 or 2
  const float* Arow = A + (size_t)row * K;
  v8f acc = {0.f, 0.f, 0.f, 0.f, 0.f, 0.f, 0.f, 0.f};
#pragma unroll
  for (int k0 = 0; k0 < K; k0 += 4) {
    int ka = k0 + khalf;
    v2f a = *(const v2f*)(Arow + ka);         // aligned b64 load
    a.x *= am; a.y *= am;
    v2f b;
    b.x = B[ka * LDB + col] * bm;
    b.y = B[(ka + 1) * LDB + col] * bm;
    acc = __builtin_amdgcn_wmma_f32_16x16x4_f32(false, a, false, b,
                                                (short)0, acc, false, false);
  }
  int rbase = (wave << 4) + ((lane >> 4) << 3);
#pragma unroll
  for (int v = 0; v < 8; ++v) {
    int r = rbase + v;
    if (r < M) D[(size_t)r * 16 + ccol] = acc[v];
  }
}

// ---------------- degree / rsqrt ----------------
__global__ void k_deg1(const int* __restrict__ dst, float* __restrict__ deg) {
  int e = blockIdx.x * blockDim.x + threadIdx.x;
  if (e < EE) atomAddF(&deg[dst[e]], 1.0f);
}
__global__ void k_deg2(const int* __restrict__ src, const int* __restrict__ dst,
                       const int* __restrict__ nidx, float* __restrict__ deg) {
  int e = blockIdx.x * blockDim.x + threadIdx.x;
  if (e >= EE) return;
  int ns = nidx[src[e]], nd = nidx[dst[e]];
  if (ns >= 0 && nd >= 0) atomAddF(&deg[nd], 1.0f);
}
__global__ void k_rsqrt(float* p, int n) {
  int i = blockIdx.x * blockDim.x + threadIdx.x;
  if (i < n) p[i] = rsqrtf(p[i]);
}

// ---------------- edge aggregations (16 feature cols) ----------------
__global__ void k_agg_norm1(const int* __restrict__ src, const int* __restrict__ dst,
                            const float* __restrict__ h, const float* __restrict__ dinv,
                            float* __restrict__ agg) {
  int e = blockIdx.x * blockDim.x + threadIdx.x;
  if (e >= EE) return;
  int s = src[e], d = dst[e];
  float nrm = dinv[s] * dinv[d];
  const float4* hp = (const float4*)(h + (size_t)s * 16);
  float* ap = agg + (size_t)d * 16;
#pragma unroll
  for (int q = 0; q < 4; ++q) {
    float4 v = hp[q];
    atomAddF(ap + 4 * q + 0, v.x * nrm);
    atomAddF(ap + 4 * q + 1, v.y * nrm);
    atomAddF(ap + 4 * q + 2, v.z * nrm);
    atomAddF(ap + 4 * q + 3, v.w * nrm);
  }
}
__global__ void k_agg_plain1(const int* __restrict__ src, const int* __restrict__ dst,
                             const float* __restrict__ h, float* __restrict__ agg) {
  int e = blockIdx.x * blockDim.x + threadIdx.x;
  if (e >= EE) return;
  int s = src[e], d = dst[e];
  const float4* hp = (const float4*)(h + (size_t)s * 16);
  float* ap = agg + (size_t)d * 16;
#pragma unroll
  for (int q = 0; q < 4; ++q) {
    float4 v = hp[q];
    atomAddF(ap + 4 * q + 0, v.x);
    atomAddF(ap + 4 * q + 1, v.y);
    atomAddF(ap + 4 * q + 2, v.z);
    atomAddF(ap + 4 * q + 3, v.w);
  }
}
__global__ void k_agg_norm2(const int* __restrict__ src, const int* __restrict__ dst,
                            const int* __restrict__ nidx, const float* __restrict__ h,
                            const float* __restrict__ dinv, float* __restrict__ agg) {
  int e = blockIdx.x * blockDim.x + threadIdx.x;
  if (e >= EE) return;
  int ns = nidx[src[e]], nd = nidx[dst[e]];
  if (ns < 0 || nd < 0) return;
  float nrm = dinv[ns] * dinv[nd];
  const float4* hp = (const float4*)(h + (size_t)ns * 16);
  float* ap = agg + (size_t)nd * 16;
#pragma unroll
  for (int q = 0; q < 4; ++q) {
    float4 v = hp[q];
    atomAddF(ap + 4 * q + 0, v.x * nrm);
    atomAddF(ap + 4 * q + 1, v.y * nrm);
    atomAddF(ap + 4 * q + 2, v.z * nrm);
    atomAddF(ap + 4 * q + 3, v.w * nrm);
  }
}
__global__ void k_agg_plain2(const int* __restrict__ src, const int* __restrict__ dst,
                             const int* __restrict__ nidx, const float* __restrict__ h,
                             float* __restrict__ agg) {
  int e = blockIdx.x * blockDim.x + threadIdx.x;
  if (e >= EE) return;
  int ns = nidx[src[e]], nd = nidx[dst[e]];
  if (ns < 0 || nd < 0) return;
  const float4* hp = (const float4*)(h + (size_t)ns * 16);
  float* ap = agg + (size_t)nd * 16;
#pragma unroll
  for (int q = 0; q < 4; ++q) {
    float4 v = hp[q];
    atomAddF(ap + 4 * q + 0, v.x);
    atomAddF(ap + 4 * q + 1, v.y);
    atomAddF(ap + 4 * q + 2, v.z);
    atomAddF(ap + 4 * q + 3, v.w);
  }
}

// ---------------- finalize conv (self-loop + bias [+relu]) ----------------
__global__ void k_fin1(const float* __restrict__ h0, const float* __restrict__ agg,
                       const float* __restrict__ dinv, const float* __restrict__ b,
                       float* __restrict__ out, int n) {
  int idx = blockIdx.x * blockDim.x + threadIdx.x;
  if (idx >= n * 16) return;
  int i = idx >> 4, c = idx & 15;
  float di = dinv[i];
  float v = agg[idx] + h0[idx] * di * di + b[c];
  out[idx] = fmaxf(v, 0.f);
}
__global__ void k_fin2(const float* __restrict__ hp, const float* __restrict__ agg,
                       const float* __restrict__ dinv, const float* __restrict__ b,
                       float* __restrict__ out, int n, int C) {
  int idx = blockIdx.x * blockDim.x + threadIdx.x;
  if (idx >= n * 16) return;
  int i = idx >> 4, c = idx & 15;
  float di = dinv[i];
  float v = (c < C) ? (agg[idx] + hp[idx] * di * di + b[c]) : 0.f;
  out[idx] = v;
}

// ---------------- SAGPool score ----------------
__global__ void k_score(const float* __restrict__ x, const float* __restrict__ agg,
                        const float* __restrict__ wroot, const float* __restrict__ wrel,
                        const float* __restrict__ bias, float* __restrict__ score,
                        int n, int dim) {
  int i = blockIdx.x * blockDim.x + threadIdx.x;
  if (i >= n) return;
  float s = bias[0];
  for (int c = 0; c < dim; ++c)
    s += x[(size_t)i * 16 + c] * wroot[c] + agg[(size_t)i * 16 + c] * wrel[c];
  score[i] = s;
}

// ---------------- exact top-k via 4-round radix select ----------------
__device__ __forceinline__ unsigned keyOf(float f) {
  unsigned u = __float_as_uint(f);
  return u ^ ((u >> 31) ? 0xFFFFFFFFu : 0x80000000u);  // order-preserving
}
__global__ void k_sel_init(unsigned* sel, unsigned* hist, unsigned k) {
  int i = threadIdx.x;
  if (i < 1024) hist[i] = 0u;
  if (i == 0) { sel[0] = 0u; sel[1] = k; sel[2] = 0u; sel[3] = 0u; }
}
__global__ void k_sel_hist(const float* __restrict__ score, int n,
                           const unsigned* __restrict__ sel, unsigned* __restrict__ hist,
                           int round) {
  int i = blockIdx.x * blockDim.x + threadIdx.x;
  if (i >= n) return;
  unsigned key  = keyOf(score[i]);
  unsigned mask = round ? (0xFFFFFFFFu << (32 - 8 * round)) : 0u;
  int shift = 24 - 8 * round;
  if ((key & mask) == sel[0])
    atomicAdd(&hist[round * 256 + ((key >> shift) & 255u)], 1u);
}
__global__ void k_sel_scan(unsigned* sel, const unsigned* hist, int round) {
  int shift = 24 - 8 * round;
  unsigned remaining = sel[1], prefix = sel[0], cum = 0u;
  for (int b = 255; b >= 0; --b) {
    unsigned c = hist[round * 256 + b];
    if (cum + c >= remaining) {
      sel[0] = prefix | ((unsigned)b << shift);
      sel[1] = remaining - cum;
      return;
    }
    cum += c;
  }
  sel[1] = 0u;
}
__global__ void k_sel_compact(const float* __restrict__ score, int n,
                              unsigned* __restrict__ sel, int* __restrict__ perm, int k) {
  int i = blockIdx.x * blockDim.x + threadIdx.x;
  if (i >= n) return;
  unsigned key = keyOf(score[i]);
  unsigned T = sel[0];
  if (key > T) {
    unsigned p = atomicAdd(&sel[2], 1u);
    perm[p] = i;
  } else if (key == T) {
    unsigned e = atomicAdd(&sel[3], 1u);
    unsigned quota = sel[1];
    if (e < quota) perm[(unsigned)k - quota + e] = i;
  }
}

// ---------------- gather kept nodes: x_out[r] = x[perm[r]] * tanh(score) ----
__global__ void k_gather(const float* __restrict__ x, const float* __restrict__ score,
                         const int* __restrict__ perm, int k, float* __restrict__ out,
                         int* nidx) {
  int r = blockIdx.x * blockDim.x + threadIdx.x;
  if (r >= k) return;
  int p = perm[r];
  float t = tanhf(score[p]);
  const float4* xp = (const float4*)(x + (size_t)p * 16);
  float4* op = (float4*)(out + (size_t)r * 16);
#pragma unroll
  for (int q = 0; q < 4; ++q) {
    float4 v = xp[q];
    op[q] = make_float4(v.x * t, v.y * t, v.z * t, v.w * t);
  }
  if (nidx) nidx[p] = r;
}

// ---------------- mean over rows + log_softmax ----------------
__global__ void k_final(const float* __restrict__ x2, int k2, float* __restrict__ out) {
  __shared__ float red[256][CC];
  int tid = threadIdx.x;
  float s[CC];
#pragma unroll
  for (int c = 0; c < CC; ++c) s[c] = 0.f;
  for (int r = tid; r < k2; r += 256)
#pragma unroll
    for (int c = 0; c < CC; ++c) s[c] += x2[(size_t)r * 16 + c];
#pragma unroll
  for (int c = 0; c < CC; ++c) red[tid][c] = s[c];
  __syncthreads();
  for (int off = 128; off > 0; off >>= 1) {
    if (tid < off)
#pragma unroll
      for (int c = 0; c < CC; ++c) red[tid][c] += red[tid + off][c];
    __syncthreads();
  }
  if (tid == 0) {
    float v[CC], m = -INFINITY;
#pragma unroll
    for (int c = 0; c < CC; ++c) { v[c] = red[0][c] / (float)k2; m = fmaxf(m, v[c]); }
    float lse = 0.f;
#pragma unroll
    for (int c = 0; c < CC; ++c) lse += expf(v[c] - m);
    lse = logf(lse);
#pragma unroll
    for (int c = 0; c < CC; ++c) out[c] = v[c] - m - lse;
  }
}

static inline int cdiv(int a, int b) { return (a + b - 1) / b; }

extern "C" void kernel_launch(void* const* d_in, const int* in_sizes, int n_in,
                              void* d_out, int out_size, void* d_ws, size_t ws_size,
                              hipStream_t stream) {
  (void)in_sizes; (void)n_in; (void)out_size; (void)ws_size;
  const float* x        = (const float*)d_in[0];
  const int*   esrc     = (const int*)  d_in[1];
  const int*   edst     = (const int*)  d_in[2];
  const float* W1       = (const float*)d_in[3];
  const float* b1       = (const float*)d_in[4];
  const float* p1_wroot = (const float*)d_in[5];
  const float* p1_wrel  = (const float*)d_in[6];
  const float* p1_b     = (const float*)d_in[7];
  const float* W2       = (const float*)d_in[8];
  const float* b2       = (const float*)d_in[9];
  const float* p2_wroot = (const float*)d_in[10];
  const float* p2_wrel  = (const float*)d_in[11];
  const float* p2_b     = (const float*)d_in[12];
  float* out = (float*)d_out;

  // ---- workspace layout (~35.5 MB) ----
  float* ws     = (float*)d_ws;
  float* h0     = ws;                 // NN*HH  (x@W1)
  float* agg1   = h0    + NN * HH;    // NN*HH
  float* hbuf   = agg1  + NN * HH;    // NN*HH  (post-conv1 relu)
  float* aggs1  = hbuf  + NN * HH;    // NN*HH  (pool1 score agg)
  float* deg1   = aggs1 + NN * HH;    // NN     (deg -> dinv in place)
  float* score1 = deg1  + NN;         // NN
  float* x1     = score1 + NN;        // KK1*HH
  float* h2pre  = x1    + KK1 * HH;   // KK1*HH (x1@W2, cols 10..15 == 0)
  float* agg2   = h2pre + KK1 * HH;   // KK1*HH
  float* h2     = agg2  + KK1 * HH;   // KK1*HH
  float* aggs2  = h2    + KK1 * HH;   // KK1*HH
  float* deg2   = aggs2 + KK1 * HH;   // KK1
  float* score2 = deg2  + KK1;        // KK1
  float* x2     = score2 + KK1;       // KK2*HH
  int*   perm1  = (int*)(x2 + KK2 * HH);  // KK1
  int*   nidx1  = perm1 + KK1;            // NN
  int*   perm2  = nidx1 + NN;             // KK2
  unsigned* hist = (unsigned*)(perm2 + KK2);  // 4*256
  unsigned* sel  = hist + 1024;               // prefix, remaining, c_gt, c_eq

  const int TB = 256;
  const int EG = cdiv(EE, TB);

  // ---- re-init accumulators every call (graph-replay safe) ----
  k_fill_f32<<<cdiv(NN * HH, TB), TB, 0, stream>>>(agg1, 0.f, NN * HH);
  k_fill_f32<<<cdiv(NN * HH, TB), TB, 0, stream>>>(aggs1, 0.f, NN * HH);
  k_fill_f32<<<cdiv(NN, TB), TB, 0, stream>>>(deg1, 1.f, NN);   // +1 self loop
  k_fill_f32<<<cdiv(KK1 * HH, TB), TB, 0, stream>>>(agg2, 0.f, KK1 * HH);
  k_fill_f32<<<cdiv(KK1 * HH, TB), TB, 0, stream>>>(aggs2, 0.f, KK1 * HH);
  k_fill_f32<<<cdiv(KK1, TB), TB, 0, stream>>>(deg2, 1.f, KK1);
  k_fill_i32<<<cdiv(NN, TB), TB, 0, stream>>>(nidx1, -1, NN);

  // ---- conv1: h = relu(GCNConv(x)) ----
  k_gemm16<FF, HH, HH><<<cdiv(cdiv(NN, 16), 4), 128, 0, stream>>>(x, W1, h0, NN);
  k_deg1<<<EG, TB, 0, stream>>>(edst, deg1);
  k_rsqrt<<<cdiv(NN, TB), TB, 0, stream>>>(deg1, NN);
  k_agg_norm1<<<EG, TB, 0, stream>>>(esrc, edst, h0, deg1, agg1);
  k_fin1<<<cdiv(NN * HH, TB), TB, 0, stream>>>(h0, agg1, deg1, b1, hbuf, NN);

  // ---- pool1 ----
  k_agg_plain1<<<EG, TB, 0, stream>>>(esrc, edst, hbuf, aggs1);
  k_score<<<cdiv(NN, TB), TB, 0, stream>>>(hbuf, aggs1, p1_wroot, p1_wrel, p1_b,
                                           score1, NN, HH);
  k_sel_init<<<1, 1024, 0, stream>>>(sel, hist, (unsigned)KK1);
  for (int r = 0; r < 4; ++r) {
    k_sel_hist<<<cdiv(NN, TB), TB, 0, stream>>>(score1, NN, sel, hist, r);
    k_sel_scan<<<1, 1, 0, stream>>>(sel, hist, r);
  }
  k_sel_compact<<<cdiv(NN, TB), TB, 0, stream>>>(score1, NN, sel, perm1, KK1);
  k_gather<<<cdiv(KK1, TB), TB, 0, stream>>>(hbuf, score1, perm1, KK1, x1, nidx1);

  // ---- conv2 (edges remapped on the fly through nidx1; ew==0 == dropped) ----
  k_gemm16<HH, CC, CC><<<cdiv(cdiv(KK1, 16), 4), 128, 0, stream>>>(x1, W2, h2pre, KK1);
  k_deg2<<<EG, TB, 0, stream>>>(esrc, edst, nidx1, deg2);
  k_rsqrt<<<cdiv(KK1, TB), TB, 0, stream>>>(deg2, KK1);
  k_agg_norm2<<<EG, TB, 0, stream>>>(esrc, edst, nidx1, h2pre, deg2, agg2);
  k_fin2<<<cdiv(KK1 * HH, TB), TB, 0, stream>>>(h2pre, agg2, deg2, b2, h2, KK1, CC);

  // ---- pool2 ----
  k_agg_plain2<<<EG, TB, 0, stream>>>(esrc, edst, nidx1, h2, aggs2);
  k_score<<<cdiv(KK1, TB), TB, 0, stream>>>(h2, aggs2, p2_wroot, p2_wrel, p2_b,
                                            score2, KK1, CC);
  k_sel_init<<<1, 1024, 0, stream>>>(sel, hist, (unsigned)KK2);
  for (int r = 0; r < 4; ++r) {
    k_sel_hist<<<cdiv(KK1, TB), TB, 0, stream>>>(score2, KK1, sel, hist, r);
    k_sel_scan<<<1, 1, 0, stream>>>(sel, hist, r);
  }
  k_sel_compact<<<cdiv(KK1, TB), TB, 0, stream>>>(score2, KK1, sel, perm2, KK2);
  k_gather<<<cdiv(KK2, TB), TB, 0, stream>>>(h2, score2, perm2, KK2, x2, nullptr);

  // ---- global mean pool + log_softmax -> d_out[10] ----
  k_final<<<1, 256, 0, stream>>>(x2, KK2, out);
}